// HadamardSimMultiChannel8_16ch4q_3453153706440
// MI455X (gfx1250) — compile-verified
//
#include <hip/hip_runtime.h>

typedef __attribute__((ext_vector_type(2))) float v2f;
typedef __attribute__((ext_vector_type(8))) float v8f;
typedef __attribute__((ext_vector_type(4))) int   v4i;

#define N_AB 8
#define NUM_CH 16
#define NPARAM_CH 72      // (M+1)*N_AB
#define NPARAM_S2 36      // (M+1)*N_S2
#define KTOT 256
#define KCH 16            // K-chunk staged through LDS
#define NCHUNK (KTOT / KCH)
#define XS_STRIDE 20      // 16 + pad, 80B row stride (16B aligned, conflict-free A-frags)
#define SW_STRIDE 260     // 256 + pad (breaks 16-way bank conflict)
#define ZB_STRIDE 20
#define ROWS_PER_BLOCK 256

// CDNA5 async global->LDS path (bypasses VGPRs, tracked by ASYNCcnt)
#if defined(__has_builtin)
#if __has_builtin(__builtin_amdgcn_global_load_async_to_lds_b128) && \
    __has_builtin(__builtin_amdgcn_s_wait_asynccnt)
#define USE_ASYNC_LDS 1
#endif
#endif

#ifdef USE_ASYNC_LDS
typedef __attribute__((address_space(1))) v4i glob_v4i;  // global b128 payload
typedef __attribute__((address_space(3))) v4i lds_v4i;   // LDS b128 payload
#endif

// ---------------------------------------------------------------------------
// nan_to_num(x): nan -> 0, +-inf -> 0 (matches jnp.nan_to_num(posinf=0,neginf=0))
__device__ __forceinline__ float n2n(float v) {
    if (v != v) return 0.0f;
    if (v > 3.0e38f || v < -3.0e38f) return 0.0f;
    return v;
}

// ---------------------------------------------------------------------------
// Kernel 1: build the 16 per-channel W states (8-qubit statevector sim in LDS)
// and the stage-2 cos/sin table.  Grid = 16 blocks x 256 threads.
__global__ __launch_bounds__(256)
void build_states_kernel(const float* __restrict__ theta_ch,
                         const float* __restrict__ theta_s2,
                         float* __restrict__ sw,   // [16][256]
                         float* __restrict__ cs) { // [36][2]
    __shared__ float st[256];
    const int t  = threadIdx.x;
    const int ch = blockIdx.x;

    // stage-2 cos/sin table (uniform across batch) - block 0 only
    if (ch == 0 && t < NPARAM_S2) {
        float a = 0.5f * theta_s2[t];
        cs[2 * t]     = cosf(a);
        cs[2 * t + 1] = sinf(a);
    }

    const float* th = theta_ch + ch * NPARAM_CH;
    st[t] = (t == 0) ? 1.0f : 0.0f;
    __syncthreads();

    int p = 0;
    for (int m = 0; m < 8; ++m) {
        // RY layer: qubit q is the q-th MSB -> bit position 7-q (LSB=0)
        for (int q = 0; q < N_AB; ++q) {
            float a = 0.5f * th[p++];
            float c = cosf(a), sn = sinf(a);
            int bq = 7 - q, mask = 1 << bq;
            float v0 = st[t & ~mask];
            float v1 = st[t |  mask];
            __syncthreads();
            st[t] = ((t >> bq) & 1) ? (sn * v0 + c * v1) : (c * v0 - sn * v1);
            __syncthreads();
        }
        // circular CNOTs (control q -> target tgt): permutation of amplitudes
        for (int q = 0; q < N_AB; ++q) {
            int tgt = (m & 1) ? ((q + 7) & 7) : ((q + 1) & 7);
            int bc = 7 - q, bt = 7 - tgt;
            float v = ((t >> bc) & 1) ? st[t ^ (1 << bt)] : st[t];
            __syncthreads();
            st[t] = v;
            __syncthreads();
        }
    }
    for (int q = 0; q < N_AB; ++q) {
        float a = 0.5f * th[p++];
        float c = cosf(a), sn = sinf(a);
        int bq = 7 - q, mask = 1 << bq;
        float v0 = st[t & ~mask];
        float v1 = st[t |  mask];
        __syncthreads();
        st[t] = ((t >> bq) & 1) ? (sn * v0 + c * v1) : (c * v0 - sn * v1);
        __syncthreads();
    }
    sw[ch * KTOT + t] = st[t];
}

// ---------------------------------------------------------------------------
// Stage-2 register-resident 4-qubit gates (all indices constant-folded).
__device__ __forceinline__ void ry4(float s[16], int q, float c, float sn) {
    const int bq = 3 - q, m = 1 << bq;
#pragma unroll
    for (int d = 0; d < 16; ++d) {
        if (!((d >> bq) & 1)) {
            float a = s[d], b = s[d | m];
            s[d]     = c  * a - sn * b;
            s[d | m] = sn * a + c  * b;
        }
    }
}

__device__ __forceinline__ void cnot4(float s[16], int q1, int q2) {
    const int bc = 3 - q1, bt = 3 - q2;
#pragma unroll
    for (int d = 0; d < 16; ++d) {
        if (((d >> bc) & 1) && !((d >> bt) & 1)) {
            float tmp = s[d];
            s[d] = s[d | (1 << bt)];
            s[d | (1 << bt)] = tmp;
        }
    }
}

// ---------------------------------------------------------------------------
// Issue the async (or fallback) copy of one 256-row x 16-col chunk of X.
// Each thread moves 4x b128 (16 floats).  Async variant bypasses VGPRs and
// is tracked with ASYNCcnt (in-order completion -> s_wait_asynccnt N works
// as a "previous chunk landed" condition).
__device__ __forceinline__ void issue_chunk(const float* __restrict__ x,
                                            float* xs, size_t rowBase,
                                            int c, int t) {
#pragma unroll
    for (int i = 0; i < 4; ++i) {
        int fid = i * 256 + t;             // 0..1023 float4s
        int r = fid >> 2, g = fid & 3;
        const float* gsrc = x + (rowBase + r) * KTOT + c * KCH + g * 4;
        float* ldst = &xs[r * XS_STRIDE + g * 4];
#ifdef USE_ASYNC_LDS
        __builtin_amdgcn_global_load_async_to_lds_b128(
            (glob_v4i*)gsrc, (lds_v4i*)ldst, /*offset=*/0, /*cpol=*/0);
#else
        *(float4*)ldst = *(const float4*)gsrc;
#endif
    }
}

__device__ __forceinline__ void wait_chunk_prev_pending4() {
#ifdef USE_ASYNC_LDS
    __builtin_amdgcn_s_wait_asynccnt(4);   // the 4 just-issued may remain
#endif
}
__device__ __forceinline__ void wait_chunk_all() {
#ifdef USE_ASYNC_LDS
    __builtin_amdgcn_s_wait_asynccnt(0);
#endif
}

// ---------------------------------------------------------------------------
// Kernel 2: fused  Z = X * Sw^T  (f32 WMMA)  ->  silu/normalize  ->
// 4-qubit ansatz  ->  probabilities / logP.
// Block = 256 threads (8 waves) handles 256 batch rows; each wave owns two
// 16x16 WMMA tiles (32 rows x 16 channels).  X is double-buffered in LDS via
// async global->LDS copies; the Z buffer aliases x-buffer 0 after the GEMM.
__global__ __launch_bounds__(256)
void fused_sim_kernel(const float* __restrict__ x,
                      const float* __restrict__ sw,
                      const float* __restrict__ cs,
                      float* __restrict__ out,
                      int nrows) {
    __shared__ float smem[2 * ROWS_PER_BLOCK * XS_STRIDE + NUM_CH * SW_STRIDE];
    float* const xs0 = smem;                                   // 20480 B
    float* const xs1 = smem + ROWS_PER_BLOCK * XS_STRIDE;      // 20480 B
    float* const swl = smem + 2 * ROWS_PER_BLOCK * XS_STRIDE;  // 16640 B
    float* const zb  = smem;   // aliases xs0 (only touched after final barrier)

    const int t    = threadIdx.x;
    const int lane = t & 31;
    const int wave = t >> 5;
    const int half = lane >> 4;    // 0: lanes 0-15, 1: lanes 16-31
    const int lr   = lane & 15;
    const size_t rowBase = (size_t)blockIdx.x * ROWS_PER_BLOCK;

    // Stage Sw into LDS with padded stride (coalesced global read)
#pragma unroll
    for (int i = 0; i < 16; ++i) {
        int idx = t + i * 256;             // 0..4095
        int chn = idx >> 8, k = idx & 255;
        swl[chn * SW_STRIDE + k] = sw[idx];
    }

    // Prologue: start streaming chunk 0
    issue_chunk(x, xs0, rowBase, 0, t);

    v8f acc0 = {}; // rows wave*32 + 0..15
    v8f acc1 = {}; // rows wave*32 + 16..31
    const int rowT0 = wave * 32 + lr;      // A-frag row, tile 0 (lane's M)
    const int rowT1 = rowT0 + 16;          // tile 1

    for (int c = 0; c < NCHUNK; ++c) {
        float* const xs = (c & 1) ? xs1 : xs0;
        if (c + 1 < NCHUNK) {
            // Refill the other buffer (its last readers passed the barrier
            // at the end of iteration c-1), then wait for chunk c to land.
            issue_chunk(x, (c & 1) ? xs0 : xs1, rowBase, c + 1, t);
            wait_chunk_prev_pending4();
        } else {
            wait_chunk_all();
        }
        __syncthreads();

#pragma unroll
        for (int kk = 0; kk < KCH / 4; ++kk) {
            const int kl = kk * 4 + 2 * half;           // local K for A frag
            const int kg = c * KCH + kk * 4 + 2 * half; // global K for B frag
            // A 16x4 f32: lane l<16 -> (M=l, K=k0,k0+1); lane l+16 -> K=k0+2,k0+3
            v2f a0 = *(const v2f*)&xs[rowT0 * XS_STRIDE + kl];
            v2f a1 = *(const v2f*)&xs[rowT1 * XS_STRIDE + kl];
            // B 4x16 f32 (B[k][n] = Sw[n][k]): lane = N, K mirrored like A
            v2f bb = *(const v2f*)&swl[lr * SW_STRIDE + kg];
            acc0 = __builtin_amdgcn_wmma_f32_16x16x4_f32(
                false, a0, false, bb, (short)0, acc0, false, false);
            acc1 = __builtin_amdgcn_wmma_f32_16x16x4_f32(
                false, a1, false, bb, (short)0, acc1, false, false);
        }
        __syncthreads();   // all waves done with this buffer before refill
    }

    // Scatter Z tiles: C VGPR v -> M = v + 8*half, N = lr  (zb aliases xs0)
    {
        const int base0 = (wave * 32 + half * 8) * ZB_STRIDE + lr;
        const int base1 = base0 + 16 * ZB_STRIDE;
#pragma unroll
        for (int v = 0; v < 8; ++v) {
            zb[base0 + v * ZB_STRIDE] = acc0[v];
            zb[base1 + v * ZB_STRIDE] = acc1[v];
        }
    }
    __syncthreads();

    // ---- Stage 2: one full batch row per thread, state in registers ----
    float s[16];
#pragma unroll
    for (int i = 0; i < 4; ++i) {
        float4 v = *(const float4*)&zb[t * ZB_STRIDE + i * 4];
        s[4 * i + 0] = v.x; s[4 * i + 1] = v.y;
        s[4 * i + 2] = v.z; s[4 * i + 3] = v.w;
    }

    // silu + safe_normalize
    float nrm2 = 0.0f;
#pragma unroll
    for (int i = 0; i < 16; ++i) {
        float z = s[i];
        float f = n2n(z / (1.0f + __expf(-z)));
        s[i] = f;
        nrm2 += f * f;
    }
    float inv = 1.0f / fmaxf(sqrtf(nrm2), 1e-6f);
#pragma unroll
    for (int i = 0; i < 16; ++i) s[i] *= inv;

    // ansatz_block: 8 x (RY layer + circular CNOTs) + final RY layer
    int p = 0;
#pragma unroll
    for (int m = 0; m < 8; ++m) {
#pragma unroll
        for (int q = 0; q < 4; ++q) { ry4(s, q, cs[2 * p], cs[2 * p + 1]); ++p; }
        if ((m & 1) == 0) {
            cnot4(s, 0, 1); cnot4(s, 1, 2); cnot4(s, 2, 3); cnot4(s, 3, 0);
        } else {
            cnot4(s, 0, 3); cnot4(s, 1, 0); cnot4(s, 2, 1); cnot4(s, 3, 2);
        }
    }
#pragma unroll
    for (int q = 0; q < 4; ++q) { ry4(s, q, cs[2 * p], cs[2 * p + 1]); ++p; }

    // safe_normalize again
    float nrm2b = 0.0f;
#pragma unroll
    for (int i = 0; i < 16; ++i) { float v = n2n(s[i]); s[i] = v; nrm2b += v * v; }
    float invb = 1.0f / fmaxf(sqrtf(nrm2b), 1e-6f);
#pragma unroll
    for (int i = 0; i < 16; ++i) s[i] *= invb;

    // marginalize first qubit (MSB): P[k] = s[k]^2 + s[k+8]^2
    float P[8], psum = 0.0f;
#pragma unroll
    for (int k = 0; k < 8; ++k) {
        float pv = fmaxf(n2n(s[k] * s[k] + s[k + 8] * s[k + 8]), 0.0f);
        P[k] = pv; psum += pv;
    }
    float invp = 1.0f / fmaxf(psum, 1e-6f);
    float lg[8];
#pragma unroll
    for (int k = 0; k < 8; ++k) {
        float pv = P[k] * invp;
        P[k] = pv;
        lg[k] = __logf(fmaxf(pv, 1e-12f));
    }

    const size_t row = rowBase + t;
    float4* o0 = (float4*)(out + row * 8);                        // logp
    float4* o1 = (float4*)(out + (size_t)nrows * 8 + row * 8);    // P
    o0[0] = make_float4(lg[0], lg[1], lg[2], lg[3]);
    o0[1] = make_float4(lg[4], lg[5], lg[6], lg[7]);
    o1[0] = make_float4(P[0], P[1], P[2], P[3]);
    o1[1] = make_float4(P[4], P[5], P[6], P[7]);
}

// ---------------------------------------------------------------------------
extern "C" void kernel_launch(void* const* d_in, const int* in_sizes, int n_in,
                              void* d_out, int out_size, void* d_ws, size_t ws_size,
                              hipStream_t stream) {
    const float* x        = (const float*)d_in[0];
    const float* theta_ch = (const float*)d_in[1];
    const float* theta_s2 = (const float*)d_in[2];
    float* out = (float*)d_out;

    float* sw = (float*)d_ws;                 // 16*256 floats
    float* cs = sw + NUM_CH * KTOT;           // 72 floats

    const int nrows = in_sizes[0] / KTOT;     // 131072

    hipLaunchKernelGGL(build_states_kernel, dim3(NUM_CH), dim3(256), 0, stream,
                       theta_ch, theta_s2, sw, cs);
    hipLaunchKernelGGL(fused_sim_kernel, dim3(nrows / ROWS_PER_BLOCK), dim3(256),
                       0, stream, x, sw, cs, out, nrows);
}